// HAN_55018531062476
// MI455X (gfx1250) — compile-verified
//
#include <hip/hip_runtime.h>
#include <math.h>

typedef __attribute__((ext_vector_type(2))) float v2f;
typedef __attribute__((ext_vector_type(8))) float v8f;

#define N_BUG   200000
#define N_USER  100000
#define NEDGE   2000000
#define IN_DIM  256
#define HID     128
#define OUTC    64
#define NCLS    128
#define NEG_SLOPE 0.2f

// ---------------------------------------------------------------- utilities

__global__ void fill_kernel(float* __restrict__ p, size_t n, float v) {
  size_t i = (size_t)blockIdx.x * blockDim.x + threadIdx.x;
  if (i < n) p[i] = v;
}

__device__ __forceinline__ float leaky(float x) {
  return x > 0.0f ? x : NEG_SLOPE * x;
}

// order-preserving float atomic max (init with -inf)
__device__ __forceinline__ void atomicMaxF(float* addr, float val) {
  if (val >= 0.0f)
    atomicMax((int*)addr, __float_as_int(val));
  else
    atomicMin((unsigned int*)addr, (unsigned int)__float_as_int(val));
}

// ------------------------------------------------- fp32 WMMA tiled GEMM
// C[M,N] = act(A)[M,K] @ W[K,N] + bias[N],  act = optional ReLU on A.
// One block = 32 rows of A (two 16-row WMMA tiles), (N/16) waves, each wave
// owns one 16-col tile and both M tiles -> B fragment reused across 2 WMMAs.
// K compile-time (256/128/64); M % 32 == 0, N % 16 == 0.
template <int K, bool RELU_A>
__global__ void gemm_wmma_f32(const float* __restrict__ A,
                              const float* __restrict__ W,
                              const float* __restrict__ bias,
                              float* __restrict__ C,
                              int M, int N) {
  constexpr int KP = K + 1;                 // pad to dodge LDS bank conflicts
  __shared__ float As[32 * KP];
  const int m0 = blockIdx.x * 32;
  const int tid = threadIdx.x;

  // stage contiguous 32xK A tile into LDS (coalesced), fusing ReLU
  for (int i = tid; i < 32 * K; i += blockDim.x) {
    int row = i / K;
    int col = i - row * K;
    float v = A[(size_t)m0 * K + i];
    if (RELU_A) v = fmaxf(v, 0.0f);
    As[row * KP + col] = v;
  }
  __syncthreads();

  const int lane = tid & 31;
  const int half = lane >> 4;               // 0: K pair {0,1}, 1: K pair {2,3}
  const int r    = lane & 15;               // A row (M) / B,C col (N)
  const int n0   = (tid >> 5) * 16;         // this wave's N tile

  v8f acc0, acc1;
  const float bv = bias[n0 + r];
#pragma unroll
  for (int j = 0; j < 8; ++j) { acc0[j] = bv; acc1[j] = bv; }

  const float* A0 = &As[r * KP + 2 * half];
  const float* A1 = &As[(16 + r) * KP + 2 * half];
  const float* Wp = &W[n0 + r];

#pragma unroll 8
  for (int k0 = 0; k0 < K; k0 += 4) {
    v2f b;
    // B 4x16 layout mirrors A: v0 = row K (2*half), v1 = row K+1; lanes = N
    b.x = Wp[(k0 + 2 * half) * N];
    b.y = Wp[(k0 + 2 * half + 1) * N];
    v2f a0, a1;
    a0.x = A0[k0]; a0.y = A0[k0 + 1];
    a1.x = A1[k0]; a1.y = A1[k0 + 1];
    acc0 = __builtin_amdgcn_wmma_f32_16x16x4_f32(false, a0, false, b,
                                                 (short)0, acc0, false, false);
    acc1 = __builtin_amdgcn_wmma_f32_16x16x4_f32(false, a1, false, b,
                                                 (short)0, acc1, false, false);
  }

  // C/D layout: VGPR j -> row (tile_m + j + 8*half), col n0 + (lane&15)
#pragma unroll
  for (int j = 0; j < 8; ++j) {
    C[(size_t)(m0 + j + 8 * half) * N + n0 + r]      = acc0[j];
    C[(size_t)(m0 + 16 + j + 8 * half) * N + n0 + r] = acc1[j];
  }
}

// ------------------------------------------------- per-node attention logits
// s[n,h] = sum_d h_feat[n, h*D + d] * att[h*D + d]
__global__ void node_scores(const float* __restrict__ hfeat,
                            const float* __restrict__ att,
                            float* __restrict__ s, int N, int H, int D) {
  int idx = blockIdx.x * blockDim.x + threadIdx.x;
  if (idx >= N * H) return;
  int n = idx / H, h = idx - n * H;
  const float* row = hfeat + (size_t)n * H * D + h * D;
  const float* a = att + h * D;
  float sum = 0.0f;
  for (int d = 0; d < D; ++d) sum += row[d] * a[d];
  s[idx] = sum;
}

// ------------------------------------------------- edge softmax pass 1: max
__global__ void attn_max(const int* __restrict__ src, const int* __restrict__ dst,
                         const float* __restrict__ ssrc, const float* __restrict__ sdst,
                         float* __restrict__ m, int E, int H) {
  int e = blockIdx.x * blockDim.x + threadIdx.x;
  if (e >= E) return;
  int s = src[e], d = dst[e];
  for (int h = 0; h < H; ++h) {
    float a = leaky(ssrc[s * H + h] + sdst[d * H + h]);
    atomicMaxF(&m[d * H + h], a);
  }
}

// ------------------------------------------------- edge softmax pass 2: sum
__global__ void attn_sum(const int* __restrict__ src, const int* __restrict__ dst,
                         const float* __restrict__ ssrc, const float* __restrict__ sdst,
                         const float* __restrict__ m, float* __restrict__ z,
                         int E, int H) {
  int e = blockIdx.x * blockDim.x + threadIdx.x;
  if (e >= E) return;
  int s = src[e], d = dst[e];
  for (int h = 0; h < H; ++h) {
    float a = leaky(ssrc[s * H + h] + sdst[d * H + h]);
    atomicAdd(&z[d * H + h], __expf(a - m[d * H + h]));
  }
}

// ------------------------------------------------- edge softmax pass 3: scatter
// one wave per edge; lanes cover HD channels; per-head alpha via __shfl
__global__ void attn_scatter(const int* __restrict__ src, const int* __restrict__ dst,
                             const float* __restrict__ ssrc, const float* __restrict__ sdst,
                             const float* __restrict__ m, const float* __restrict__ z,
                             const float* __restrict__ hsrc, float* __restrict__ out,
                             int E, int H, int HD) {
  int wid = blockIdx.x * (blockDim.x >> 5) + (threadIdx.x >> 5);
  if (wid >= E) return;
  int lane = threadIdx.x & 31;
  int s = src[wid], d = dst[wid];

  const float* srow = hsrc + (size_t)s * HD;
  float* drow = out + (size_t)d * HD;
  __builtin_prefetch(&srow[lane * 4], 0, 1);   // pull gathered row while alpha computes

  float al = 0.0f;
  if (lane < H) {
    float a = leaky(ssrc[s * H + lane] + sdst[d * H + lane]);
    al = __expf(a - m[d * H + lane]) / fmaxf(z[d * H + lane], 1e-16f);
  }

  int cpl = HD >> 5;                           // channels per lane (4 or 2)
  for (int i = 0; i < cpl; ++i) {
    int c = i * 32 + lane;                     // coalesced across lanes
    int h = c >> 6;                            // D = 64 per head
    float alpha = __shfl(al, h, 32);
    atomicAdd(&drow[c], srow[c] * alpha);
  }
}

// ---------------------------------------------------------------- launcher

extern "C" void kernel_launch(void* const* d_in, const int* in_sizes, int n_in,
                              void* d_out, int out_size, void* d_ws, size_t ws_size,
                              hipStream_t stream) {
  const float* xb      = (const float*)d_in[0];
  const float* xu      = (const float*)d_in[1];
  const int*   ebu_src = (const int*)d_in[2];
  const int*   ebu_dst = (const int*)d_in[3];
  const int*   eub_src = (const int*)d_in[4];
  const int*   eub_dst = (const int*)d_in[5];
  const float* w1_bug  = (const float*)d_in[6];
  const float* b1_bug  = (const float*)d_in[7];
  const float* w1_user = (const float*)d_in[8];
  const float* b1_user = (const float*)d_in[9];
  const float* a1_bu_s = (const float*)d_in[10];
  const float* a1_bu_d = (const float*)d_in[11];
  const float* a1_ub_s = (const float*)d_in[12];
  const float* a1_ub_d = (const float*)d_in[13];
  // 14..16: k1_w, k1_b, q1 — dead (single-relation semantic attn => beta==1)
  const float* w2_bug  = (const float*)d_in[17];
  const float* b2_bug  = (const float*)d_in[18];
  const float* w2_user = (const float*)d_in[19];
  const float* b2_user = (const float*)d_in[20];
  // 21,22: a2_bu_* — dead (layer-2 user output discarded)
  const float* a2_ub_s = (const float*)d_in[23];
  const float* a2_ub_d = (const float*)d_in[24];
  // 25..27: k2_w, k2_b, q2 — dead
  const float* wc      = (const float*)d_in[28];
  const float* bc      = (const float*)d_in[29];
  float* out = (float*)d_out;

  // ---- workspace layout (floats)
  float* ws = (float*)d_ws;
  size_t off = 0;
  auto alloc = [&](size_t n) { float* p = ws + off; off += n; return p; };
  float* hb1      = alloc((size_t)N_BUG * HID);     // 25.6M
  float* hu1      = alloc((size_t)N_USER * HID);    // 12.8M
  float* zb       = alloc((size_t)N_BUG * HID);     // layer1 bug out (pre-relu)
  float* zu       = alloc((size_t)N_USER * HID);    // layer1 user out (pre-relu)
  float* s_bu_src = alloc((size_t)N_BUG * 2);
  float* s_bu_dst = alloc((size_t)N_USER * 2);
  float* s_ub_src = alloc((size_t)N_USER * 2);
  float* s_ub_dst = alloc((size_t)N_BUG * 2);
  float* m_u = alloc((size_t)N_USER * 2);
  float* z_u = alloc((size_t)N_USER * 2);
  float* m_b = alloc((size_t)N_BUG * 2);
  float* z_b = alloc((size_t)N_BUG * 2);
  float* zb2 = alloc((size_t)N_BUG * OUTC);         // 12.8M (pre-relu)
  // layer-2 reuse (serial stream ordering makes this safe)
  float* hb2 = hb1;            // [N_BUG, 64]
  float* hu2 = hu1;            // [N_USER, 64]
  float* s2_src = s_ub_src;    // [N_USER]
  float* s2_dst = s_ub_dst;    // [N_BUG]
  float* m2 = m_b;             // [N_BUG]
  float* z2 = z_b;             // [N_BUG]

  auto fill = [&](float* p, size_t n, float v) {
    fill_kernel<<<dim3((unsigned)((n + 255) / 256)), 256, 0, stream>>>(p, n, v);
  };

  const int EB = (NEDGE + 255) / 256;          // thread-per-edge blocks
  const int SB = NEDGE / 8;                    // wave-per-edge blocks (256 thr = 8 waves)

  // ---------------- Layer 1: projections (heads=2, D=64), K=256, no input relu
  gemm_wmma_f32<IN_DIM, false><<<dim3(N_BUG / 32), dim3((HID / 16) * 32), 0, stream>>>(
      xb, w1_bug, b1_bug, hb1, N_BUG, HID);
  gemm_wmma_f32<IN_DIM, false><<<dim3(N_USER / 32), dim3((HID / 16) * 32), 0, stream>>>(
      xu, w1_user, b1_user, hu1, N_USER, HID);

  node_scores<<<dim3((N_BUG * 2 + 255) / 256), 256, 0, stream>>>(hb1, a1_bu_s, s_bu_src, N_BUG, 2, 64);
  node_scores<<<dim3((N_USER * 2 + 255) / 256), 256, 0, stream>>>(hu1, a1_bu_d, s_bu_dst, N_USER, 2, 64);
  node_scores<<<dim3((N_USER * 2 + 255) / 256), 256, 0, stream>>>(hu1, a1_ub_s, s_ub_src, N_USER, 2, 64);
  node_scores<<<dim3((N_BUG * 2 + 255) / 256), 256, 0, stream>>>(hb1, a1_ub_d, s_ub_dst, N_BUG, 2, 64);

  fill(m_u, (size_t)N_USER * 2, -INFINITY);
  fill(z_u, (size_t)N_USER * 2, 0.0f);
  fill(m_b, (size_t)N_BUG * 2, -INFINITY);
  fill(z_b, (size_t)N_BUG * 2, 0.0f);
  fill(zu, (size_t)N_USER * HID, 0.0f);
  fill(zb, (size_t)N_BUG * HID, 0.0f);

  // relation bug->user (dst = user)
  attn_max<<<dim3(EB), 256, 0, stream>>>(ebu_src, ebu_dst, s_bu_src, s_bu_dst, m_u, NEDGE, 2);
  attn_sum<<<dim3(EB), 256, 0, stream>>>(ebu_src, ebu_dst, s_bu_src, s_bu_dst, m_u, z_u, NEDGE, 2);
  attn_scatter<<<dim3(SB), 256, 0, stream>>>(ebu_src, ebu_dst, s_bu_src, s_bu_dst, m_u, z_u,
                                             hb1, zu, NEDGE, 2, HID);
  // relation user->bug (dst = bug)
  attn_max<<<dim3(EB), 256, 0, stream>>>(eub_src, eub_dst, s_ub_src, s_ub_dst, m_b, NEDGE, 2);
  attn_sum<<<dim3(EB), 256, 0, stream>>>(eub_src, eub_dst, s_ub_src, s_ub_dst, m_b, z_b, NEDGE, 2);
  attn_scatter<<<dim3(SB), 256, 0, stream>>>(eub_src, eub_dst, s_ub_src, s_ub_dst, m_b, z_b,
                                             hu1, zb, NEDGE, 2, HID);

  // ---------------- Layer 2 projections, K=128, ReLU fused on A (elu∘relu == relu)
  gemm_wmma_f32<HID, true><<<dim3(N_BUG / 32), dim3((OUTC / 16) * 32), 0, stream>>>(
      zb, w2_bug, b2_bug, hb2, N_BUG, OUTC);
  gemm_wmma_f32<HID, true><<<dim3(N_USER / 32), dim3((OUTC / 16) * 32), 0, stream>>>(
      zu, w2_user, b2_user, hu2, N_USER, OUTC);

  node_scores<<<dim3((N_USER + 255) / 256), 256, 0, stream>>>(hu2, a2_ub_s, s2_src, N_USER, 1, 64);
  node_scores<<<dim3((N_BUG + 255) / 256), 256, 0, stream>>>(hb2, a2_ub_d, s2_dst, N_BUG, 1, 64);

  fill(m2, (size_t)N_BUG, -INFINITY);
  fill(z2, (size_t)N_BUG, 0.0f);
  fill(zb2, (size_t)N_BUG * OUTC, 0.0f);

  attn_max<<<dim3(EB), 256, 0, stream>>>(eub_src, eub_dst, s2_src, s2_dst, m2, NEDGE, 1);
  attn_sum<<<dim3(EB), 256, 0, stream>>>(eub_src, eub_dst, s2_src, s2_dst, m2, z2, NEDGE, 1);
  attn_scatter<<<dim3(SB), 256, 0, stream>>>(eub_src, eub_dst, s2_src, s2_dst, m2, z2,
                                             hu2, zb2, NEDGE, 1, OUTC);

  // ---------------- classifier: out = relu(zb2) @ wc + bc -> [N_BUG, NCLS], K=64
  gemm_wmma_f32<OUTC, true><<<dim3(N_BUG / 32), dim3((NCLS / 16) * 32), 0, stream>>>(
      zb2, wc, bc, out, N_BUG, NCLS);
}